// AECOLS_71511205479085
// MI455X (gfx1250) — compile-verified
//
#include <hip/hip_runtime.h>
#include <hip/hip_bf16.h>

typedef __attribute__((ext_vector_type(2))) float v2f;
typedef __attribute__((ext_vector_type(8))) float v8f;

#define TWO_PI 6.28318530717958647692f

namespace {

constexpr int BB = 128;
constexpr int FF = 16;
constexpr int WW = 4096;

// complex-element offsets inside d_out (complex64 flattened as float2)
constexpr int OUT_BASE  = 0;                         // out : (B,W)
constexpr int E_BASE    = BB * WW;                   // e   : (B,W)
constexpr int YAA_BASE  = 2 * BB * WW;               // y_aa: (B,W)
constexpr int GRAD_BASE = 3 * BB * WW;               // grad: (B,F,W)
constexpr int LOSS_F32  = 2 * (GRAD_BASE + BB * FF * WW); // float index of loss[B]

__device__ __forceinline__ v8f wmma_f32(v2f a, v2f b, v8f c) {
  // V_WMMA_F32_16X16X4_F32: D = A(16x4) * B(4x16) + C(16x16)
  return __builtin_amdgcn_wmma_f32_16x16x4_f32(false, a, false, b, (short)0, c,
                                               false, false);
}

// One radix-16 stage over the 4096-point row held in LDS.
// 256 independent 16-point DFT sets; 16 sets per WMMA tile; 8 waves x 2 tiles.
// SIGMA=+1: inverse transform direction; SIGMA=-1: forward.
// DIT=false: DIF stage (twiddle on output). DIT=true: twiddle on input.
template <int L, int SIGMA, bool DIT>
__device__ __forceinline__ void dft16_stage(float2* sdata, int wave, int lane,
                                            const v2f br[4], const v2f bi[4],
                                            const v2f bin[4]) {
  constexpr int R = L / 16;
  const int lhalf = (lane >= 16) ? 1 : 0;
  const int Mrow  = lane & 15;         // A-matrix row owned by this lane
  const int n     = lane & 15;         // D/B-matrix column owned by this lane
  const int kofs  = 2 * lhalf;         // A-matrix K offset per ISA layout
  const float wq  = (float)SIGMA * (TWO_PI / (float)L);

  #pragma unroll
  for (int tt = 0; tt < 2; ++tt) {
    const int tile = wave * 2 + tt;

    // ---- gather A (16 vectors x 16 elements) from LDS, per ISA f32 layout ----
    const int g    = tile * 16 + Mrow;     // set id of this lane's row
    const int blk  = g / R;
    const int j    = g % R;
    const int base = blk * L + j;

    v2f ar[4], ai[4];
    #pragma unroll
    for (int c = 0; c < 4; ++c) {
      #pragma unroll
      for (int v = 0; v < 2; ++v) {
        const int t = 4 * c + v + kofs;    // K index
        float2 x = sdata[base + R * t];
        if (DIT) {                         // input twiddle  w_L^{sigma*j*t}
          float ang = wq * (float)(j * t);
          float sn, cs; __sincosf(ang, &sn, &cs);
          const float xr = x.x * cs - x.y * sn;
          const float xi = x.x * sn + x.y * cs;
          x.x = xr; x.y = xi;
        }
        ar[c][v] = x.x;
        ai[c][v] = x.y;
      }
    }

    // ---- complex 16x16x16 matmul via 16x v_wmma_f32_16x16x4_f32 ----
    v8f dr = {};
    v8f di = {};
    #pragma unroll
    for (int c = 0; c < 4; ++c) {
      if (SIGMA > 0) {      // B = br + i*bi
        dr = wmma_f32(ar[c], br[c],  dr);
        dr = wmma_f32(ai[c], bin[c], dr);
        di = wmma_f32(ar[c], bi[c],  di);
        di = wmma_f32(ai[c], br[c],  di);
      } else {              // B = br - i*bi  (conjugate DFT matrix)
        dr = wmma_f32(ar[c], br[c],  dr);
        dr = wmma_f32(ai[c], bi[c],  dr);
        di = wmma_f32(ar[c], bin[c], di);
        di = wmma_f32(ai[c], br[c],  di);
      }
    }

    // ---- scatter D back (in place; sets of this tile are wave-private) ----
    #pragma unroll
    for (int v = 0; v < 8; ++v) {
      const int Mo = v + 8 * lhalf;        // C/D layout: VGPR v -> rows v, v+8
      const int go = tile * 16 + Mo;
      const int b2 = go / R;
      const int j2 = go % R;
      float xr = dr[v], xi = di[v];
      if (!DIT) {                          // output twiddle  w_L^{sigma*j*s}
        float ang = wq * (float)(j2 * n);
        float sn, cs; __sincosf(ang, &sn, &cs);
        const float tr = xr * cs - xi * sn;
        const float ti = xr * sn + xi * cs;
        xr = tr; xi = ti;
      }
      sdata[b2 * L + j2 + R * n] = make_float2(xr, xi);
    }
  }
}

// ---------- Kernel A: y = sum_f w*u ; out = d - y (streaming, HBM-bound) ----
// 4 consecutive bins per thread -> B128 loads/stores on the 23.3 TB/s path.
__global__ __launch_bounds__(256) void aec_filter_kernel(
    const float* __restrict__ u_re, const float* __restrict__ u_im,
    const float* __restrict__ d_re, const float* __restrict__ d_im,
    const float* __restrict__ w_re, const float* __restrict__ w_im,
    float4* __restrict__ out_q, float4* __restrict__ y_q) {
  const int q = blockIdx.x * 256 + threadIdx.x;     // over B*W/4
  if (q >= (BB * WW) / 4) return;
  const int b  = q >> 10;                           // W/4 = 1024 quads per row
  const int kq = q & 1023;
  const size_t ub = (size_t)b * FF * (WW / 4) + kq; // in float4 units

  float4 yr = make_float4(0.f, 0.f, 0.f, 0.f);
  float4 yi = make_float4(0.f, 0.f, 0.f, 0.f);
  const float4* ur4 = (const float4*)u_re;
  const float4* ui4 = (const float4*)u_im;
  const float4* wr4 = (const float4*)w_re;
  const float4* wi4 = (const float4*)w_im;
  #pragma unroll
  for (int f = 0; f < FF; ++f) {
    const size_t o = ub + (size_t)f * (WW / 4);
    const float4 ur = ur4[o], ui = ui4[o];
    const float4 wr = wr4[o], wi = wi4[o];
    yr.x += wr.x * ur.x - wi.x * ui.x;  yi.x += wr.x * ui.x + wi.x * ur.x;
    yr.y += wr.y * ur.y - wi.y * ui.y;  yi.y += wr.y * ui.y + wi.y * ur.y;
    yr.z += wr.z * ur.z - wi.z * ui.z;  yi.z += wr.z * ui.z + wi.z * ur.z;
    yr.w += wr.w * ur.w - wi.w * ui.w;  yi.w += wr.w * ui.w + wi.w * ur.w;
  }
  const float4 dr = ((const float4*)d_re)[q];
  const float4 di = ((const float4*)d_im)[q];
  // out = d - y, interleaved complex: 4 complex = 2 float4
  out_q[OUT_BASE / 2 + 2 * q + 0] =
      make_float4(dr.x - yr.x, di.x - yi.x, dr.y - yr.y, di.y - yi.y);
  out_q[OUT_BASE / 2 + 2 * q + 1] =
      make_float4(dr.z - yr.z, di.z - yi.z, dr.w - yr.w, di.w - yi.w);
  y_q[2 * q + 0] = make_float4(yr.x, yi.x, yr.y, yi.y);
  y_q[2 * q + 1] = make_float4(yr.z, yi.z, yr.w, yi.w);
}

// ---------- Kernel B: antialias(x) = FFT(mask . IFFT(x)) via WMMA radix-16 --
// blockIdx: 2 per batch row (sig=0 -> y -> y_aa ; sig=1 -> out -> e, loss)
__global__ __launch_bounds__(256) void aec_antialias_kernel(
    const float4* __restrict__ y_q, float4* __restrict__ out_q,
    float* __restrict__ loss_f) {
  __shared__ float2 sdata[WW];      // 32 KB row buffer
  __shared__ float  sred[256];

  const int tid  = threadIdx.x;
  const int wave = tid >> 5;        // wave32
  const int lane = tid & 31;
  const int b    = blockIdx.x >> 1;
  const int sig  = blockIdx.x & 1;

  // 16-point DFT matrix (sigma=+1) in the ISA B-matrix register layout:
  // VGPR v, lane l: K = v + 2*(l>=16), N = l%16
  v2f br[4], bi[4], bin[4];
  {
    const int lhalf = (lane >= 16) ? 1 : 0;
    const int n = lane & 15;
    #pragma unroll
    for (int c = 0; c < 4; ++c) {
      #pragma unroll
      for (int v = 0; v < 2; ++v) {
        const int t = 4 * c + v + 2 * lhalf;
        const float ang = (TWO_PI / 16.0f) * (float)((t * n) & 15);
        float sn, cs; __sincosf(ang, &sn, &cs);
        br[c][v] = cs; bi[c][v] = sn; bin[c][v] = -sn;
      }
    }
  }

  // Fill row buffer with B128 global loads / ds_store_b128 (2 complex / op)
  {
    const float4* src = sig ? (out_q + OUT_BASE / 2 + b * (WW / 2))
                            : (y_q + b * (WW / 2));
    float4* s4 = (float4*)sdata;
    for (int i = tid; i < WW / 2; i += 256) s4[i] = src[i];
  }
  __syncthreads();

  // IFFT (sigma=+1), radix-16 DIF: natural freq in -> digit-reversed time out
  dft16_stage<4096, 1, false>(sdata, wave, lane, br, bi, bin); __syncthreads();
  dft16_stage< 256, 1, false>(sdata, wave, lane, br, bi, bin); __syncthreads();
  dft16_stage<  16, 1, false>(sdata, wave, lane, br, bi, bin); __syncthreads();

  // keep last hop (=W/2) time samples in scrambled order:
  // time n = digitrev(p) >= 2048  <=>  (p & 15) >= 8 ; fold in 1/N IFFT scale
  {
    float4* s4 = (float4*)sdata;
    for (int i = tid; i < WW / 2; i += 256) {
      const float s0 = (((2 * i) & 15) >= 8) ? (1.0f / 4096.0f) : 0.0f;
      const float s1 = (((2 * i + 1) & 15) >= 8) ? (1.0f / 4096.0f) : 0.0f;
      float4 x = s4[i];
      x.x *= s0; x.y *= s0; x.z *= s1; x.w *= s1;
      s4[i] = x;
    }
  }
  __syncthreads();

  // FFT (sigma=-1), radix-16 DIT: digit-reversed in -> natural freq out
  dft16_stage<  16, -1, true>(sdata, wave, lane, br, bi, bin); __syncthreads();
  dft16_stage< 256, -1, true>(sdata, wave, lane, br, bi, bin); __syncthreads();
  dft16_stage<4096, -1, true>(sdata, wave, lane, br, bi, bin); __syncthreads();

  // Drain row buffer with ds_load_b128 / B128 global stores; fold loss.
  float acc = 0.0f;
  {
    float4* dst = out_q + (sig ? (E_BASE / 2 + b * (WW / 2))
                               : (YAA_BASE / 2 + b * (WW / 2)));
    const float4* s4 = (const float4*)sdata;
    for (int i = tid; i < WW / 2; i += 256) {
      const float4 x = s4[i];
      dst[i] = x;
      acc += x.x * x.x + x.y * x.y + x.z * x.z + x.w * x.w;
    }
  }
  if (sig) {   // loss[b] = 0.5 * sum |e|^2  (block owns whole row)
    sred[tid] = acc;
    __syncthreads();
    for (int off = 128; off > 0; off >>= 1) {
      if (tid < off) sred[tid] += sred[tid + off];
      __syncthreads();
    }
    if (tid == 0) loss_f[b] = 0.5f * sred[0];
  }
}

// ---------- Kernel C: grad = conj(u) * e (streaming, HBM-bound) ------------
// 4 consecutive bins per thread -> B128 loads/stores.
__global__ __launch_bounds__(256) void aec_grad_kernel(
    const float* __restrict__ u_re, const float* __restrict__ u_im,
    float4* __restrict__ out_q) {
  const int q = blockIdx.x * 256 + threadIdx.x;     // over B*F*W/4
  if (q >= (BB * FF * WW) / 4) return;
  const int b  = q >> 14;                           // F*W/4 = 16384 quads
  const int kq = q & 1023;                          // quad within the W row
  const float4 ur = ((const float4*)u_re)[q];
  const float4 ui = ((const float4*)u_im)[q];
  const float4 e0 = out_q[E_BASE / 2 + b * (WW / 2) + 2 * kq + 0];
  const float4 e1 = out_q[E_BASE / 2 + b * (WW / 2) + 2 * kq + 1];
  // conj(u)*e = (ur*er + ui*ei) + i(ur*ei - ui*er)
  out_q[GRAD_BASE / 2 + 2 * q + 0] =
      make_float4(ur.x * e0.x + ui.x * e0.y, ur.x * e0.y - ui.x * e0.x,
                  ur.y * e0.z + ui.y * e0.w, ur.y * e0.w - ui.y * e0.z);
  out_q[GRAD_BASE / 2 + 2 * q + 1] =
      make_float4(ur.z * e1.x + ui.z * e1.y, ur.z * e1.y - ui.z * e1.x,
                  ur.w * e1.z + ui.w * e1.w, ur.w * e1.w - ui.w * e1.z);
}

}  // namespace

extern "C" void kernel_launch(void* const* d_in, const int* in_sizes, int n_in,
                              void* d_out, int out_size, void* d_ws,
                              size_t ws_size, hipStream_t stream) {
  (void)in_sizes; (void)n_in; (void)out_size; (void)ws_size;
  const float* u_re = (const float*)d_in[0];
  const float* u_im = (const float*)d_in[1];
  const float* d_re = (const float*)d_in[2];
  const float* d_im = (const float*)d_in[3];
  const float* w_re = (const float*)d_in[4];
  const float* w_im = (const float*)d_in[5];
  // d_in[6] = hop_size (=2048 = W/2, baked into the scrambled-domain mask)

  float4* out_q  = (float4*)d_out;
  float*  loss_f = (float*)d_out + LOSS_F32;
  float4* y_q    = (float4*)d_ws;   // y staging: 128*4096 complex = 4 MB

  aec_filter_kernel<<<(BB * WW / 4) / 256, 256, 0, stream>>>(
      u_re, u_im, d_re, d_im, w_re, w_im, out_q, y_q);
  aec_antialias_kernel<<<BB * 2, 256, 0, stream>>>(y_q, out_q, loss_f);
  aec_grad_kernel<<<(BB * FF * WW / 4) / 256, 256, 0, stream>>>(u_re, u_im,
                                                                out_q);
}